// Encoder_24575802868358
// MI455X (gfx1250) — compile-verified
//
#include <hip/hip_runtime.h>

typedef float v2f __attribute__((ext_vector_type(2)));
typedef float v8f __attribute__((ext_vector_type(8)));

constexpr int CHAR_EMB = 30;    // input channels / embedding dim
constexpr int N_FILT   = 30;    // depthwise multiplier
constexpr int KSIZE    = 3;
constexpr int WORD_EMB = 100;
constexpr int MAX_WLEN = 40;
constexpr int CONV_OUT = CHAR_EMB * N_FILT;          // 900
constexpr int OUT_DIM  = CONV_OUT + WORD_EMB;        // 1000
constexpr int XSTRIDE  = 53;                         // odd stride -> conflict-free LDS
constexpr int WPB      = 8;                          // waves (words) per block
constexpr int WLDS_SZ  = 2720;                       // 2700 weights + zero pad

// Fused exact-xor butterfly max steps. VOP2 max with DPP8 on src0 reads the
// operand twice (permuted + plain): one instruction per step, no copy, no
// canonicalize. Bit 3 (xor8 across 16 lanes) needs the LDS swizzle unit.
__device__ __forceinline__ float fmax_xor1(float v){
    float r;
    asm("v_max_num_f32_dpp %0, %1, %1 dpp8:[1,0,3,2,5,4,7,6]" : "=v"(r) : "v"(v));
    return r;
}
__device__ __forceinline__ float fmax_xor2(float v){
    float r;
    asm("v_max_num_f32_dpp %0, %1, %1 dpp8:[2,3,0,1,6,7,4,5]" : "=v"(r) : "v"(v));
    return r;
}
__device__ __forceinline__ float fmax_xor4(float v){
    float r;
    asm("v_max_num_f32_dpp %0, %1, %1 dpp8:[4,5,6,7,0,1,2,3]" : "=v"(r) : "v"(v));
    return r;
}
__device__ __forceinline__ float fmax_xor8(float v){
    // group-of-32 swizzle: and_mask=0x1f, or_mask=8, xor_mask=0 (lanes n<8 read lane n|8)
    int   p  = __builtin_amdgcn_ds_swizzle(__float_as_int(v), 0x11f);
    float pf = __int_as_float(p), r;
    asm("v_max_num_f32 %0, %1, %2" : "=v"(r) : "v"(v), "v"(pf));
    return r;
}

// One wave32 per word. Depthwise conv as V_WMMA_F32_16X16X4_F32 tiles:
//   A[16x4] = filters (M=filter row, K=tap, K=3 column zeroed)
//   B[4x16] = im2col of char embedding row (K=tap, N=time)
//   D[16x16] f32, max over N via 3 overlapping N-tiles {0,16,22} + butterfly.
__global__ __launch_bounds__(WPB * 32) void encoder_charcnn_kernel(
    const int*   __restrict__ char_ids,   // [NW, 40]
    const int*   __restrict__ word_ids,   // [NW]
    const float* __restrict__ char_emb,   // [102, 30]
    const float* __restrict__ conv_w,     // [900, 1, 3] -> [900,3]
    const float* __restrict__ conv_b,     // [900]
    const float* __restrict__ glove,      // [V, 100]
    float*       __restrict__ out,        // [NW, 1000]
    int n_words)
{
    __shared__ float w_lds[WLDS_SZ];                   // weights + zero tail
    __shared__ float b_lds[CONV_OUT];                  // bias
    __shared__ float x_lds[WPB][CHAR_EMB * XSTRIDE];   // char-emb image per wave

    const int tid  = threadIdx.x;
    const int wave = tid >> 5;
    const int lane = tid & 31;
    int w = blockIdx.x * WPB + wave;
    if (w >= n_words) w = n_words - 1;   // clamp (grid exact; keeps EXEC full)

    // ---- GloVe row copy: 25 coalesced float4 (issued early) ----
    {
        const int gid = word_ids[w];
        const float4* gsrc = (const float4*)(glove + (size_t)gid * WORD_EMB);
        float4*       gdst = (float4*)(out + (size_t)w * OUT_DIM + CONV_OUT);
        if (lane < WORD_EMB / 4) gdst[lane] = gsrc[lane];
    }

    // ---- stage conv weights (+zero tail) and bias (block-cooperative) ----
    for (int i = tid; i < WLDS_SZ; i += blockDim.x)
        w_lds[i] = (i < CONV_OUT * KSIZE) ? conv_w[i] : 0.0f;
    for (int i = tid; i < CONV_OUT; i += blockDim.x)
        b_lds[i] = conv_b[i];

    // ---- stage char embeddings for this word: x[e][p], pad p=40,41 with 0 ----
    float* x = x_lds[wave];
    if (lane < CHAR_EMB) {
        x[lane * XSTRIDE + MAX_WLEN]     = 0.0f;
        x[lane * XSTRIDE + MAX_WLEN + 1] = 0.0f;
    }
    const int* cw = char_ids + (size_t)w * MAX_WLEN;
    for (int p = 0; p < MAX_WLEN; ++p) {
        const int id = cw[p];                       // wave-uniform address
        if (lane < CHAR_EMB)
            x[lane * XSTRIDE + p] = char_emb[id * CHAR_EMB + lane];
    }
    __syncthreads();

    // ---- per-lane constants for WMMA operand layouts ----
    const int  half  = lane >> 4;     // 0: K=0,1 ; 1: K=2,3
    const int  half2 = half * 2;
    const int  n     = lane & 15;     // A: M row ; B/D: N column
    const bool s1 = (n & 1), s2 = (n & 2), s4 = (n & 4);

    for (int e = 0; e < CHAR_EMB; ++e) {
        const float* xe = x + e * XSTRIDE;
        const int bk = n + half2;
        // B tiles shared by both M-tiles; K=3 row is don't-care (finite),
        // because A's K=3 column is zeroed below. Max index bk+23 = 40 (zeroed).
        v2f b0, b1, b2;
        b0.x = xe[bk +  0]; b0.y = xe[bk +  1];
        b1.x = xe[bk + 16]; b1.y = xe[bk + 17];
        b2.x = xe[bk + 22]; b2.y = xe[bk + 23];

        #pragma unroll
        for (int mt = 0; mt < 2; ++mt) {
            // A tile: rows >= 30 read neighbor-group weights (finite garbage,
            // masked at store); w_lds tail zero-padded for e=29 overrun.
            const int ch = e * N_FILT + mt * 16 + n;
            const int wa = ch * KSIZE + half2;
            v2f a;
            a.x = w_lds[wa];
            const float ahi = w_lds[wa + 1];
            a.y = half ? 0.0f : ahi;              // zero K=3 column

            v8f cz = {};
            v8f d0 = __builtin_amdgcn_wmma_f32_16x16x4_f32(
                false, a, false, b0, (short)0, cz, false, false);
            v8f d1 = __builtin_amdgcn_wmma_f32_16x16x4_f32(
                false, a, false, b1, (short)0, cz, false, false);
            v8f d2 = __builtin_amdgcn_wmma_f32_16x16x4_f32(
                false, a, false, b2, (short)0, cz, false, false);

            // merge 3 N-tiles (v_max3), rows r / r+8 per D layout
            float m[8];
            #pragma unroll
            for (int r = 0; r < 8; ++r)
                m[r] = fmaxf(fmaxf(d0[r], d1[r]), d2[r]);

            // pack-while-reduce butterfly over the 16 N-lanes:
            // after each exact-xor reduction of lane-bit b, fold register
            // pairs selecting on bit b (row index migrates into lane bits).
            #pragma unroll
            for (int r = 0; r < 8; ++r) m[r] = fmax_xor1(m[r]);
            float p0 = s1 ? m[1] : m[0];
            float p1 = s1 ? m[3] : m[2];
            float p2 = s1 ? m[5] : m[4];
            float p3 = s1 ? m[7] : m[6];
            p0 = fmax_xor2(p0); p1 = fmax_xor2(p1);
            p2 = fmax_xor2(p2); p3 = fmax_xor2(p3);
            float q0 = s2 ? p1 : p0;
            float q1 = s2 ? p3 : p2;
            q0 = fmax_xor4(q0); q1 = fmax_xor4(q1);
            float s = s4 ? q1 : q0;
            s = fmax_xor8(s);
            // lane now holds max of row (n&7); rows = mt*16 + half*8 + n

            const int row = mt * 16 + half * 8 + n;
            if (n < 8 && row < N_FILT) {          // one coalesced store / tile
                const int c = e * N_FILT + row;
                out[(size_t)w * OUT_DIM + c] = s + b_lds[c];
            }
        }
    }
}

extern "C" void kernel_launch(void* const* d_in, const int* in_sizes, int n_in,
                              void* d_out, int out_size, void* d_ws, size_t ws_size,
                              hipStream_t stream) {
    const int*   char_ids = (const int*)  d_in[0];
    const int*   word_ids = (const int*)  d_in[1];
    const float* char_emb = (const float*)d_in[2];
    const float* conv_w   = (const float*)d_in[3];
    const float* conv_b   = (const float*)d_in[4];
    const float* glove    = (const float*)d_in[5];
    float*       out      = (float*)      d_out;

    const int n_words = in_sizes[1];                 // B*S = 8192
    const int blocks  = (n_words + WPB - 1) / WPB;
    encoder_charcnn_kernel<<<blocks, WPB * 32, 0, stream>>>(
        char_ids, word_ids, char_emb, conv_w, conv_b, glove, out, n_words);
}